// GRUSet2Set_62294205661434
// MI455X (gfx1250) — compile-verified
//
#include <hip/hip_runtime.h>
#include <hip/hip_bf16.h>
#include <math.h>

#define NB 1000000
#define BB 4096
#define DD 128
#define TT 3

typedef __attribute__((ext_vector_type(2))) float v2f;
typedef __attribute__((ext_vector_type(8))) float v8f;

__global__ void zero_kernel(float* __restrict__ p, int n) {
  int i = blockIdx.x * blockDim.x + threadIdx.x;
  if (i < n) p[i] = 0.f;
}

// batch is sorted; seg[b] = first node of graph b, seg[BB] = NB.
__global__ void seg_bounds_kernel(const int* __restrict__ batch, int* __restrict__ seg) {
  int n = blockIdx.x * blockDim.x + threadIdx.x;
  if (n > NB) return;
  if (n < NB) {
    int bn = batch[n];
    int bp = (n == 0) ? -1 : batch[n - 1];
    for (int b = bp + 1; b <= bn; ++b) seg[b] = n;
  } else {
    int bp = batch[NB - 1];
    for (int b = bp + 1; b <= BB; ++b) seg[b] = NB;
  }
}

// out[M x NC] = A[M x K] @ W[NC x K]^T + bias[NC], f32 WMMA 16x16x4.
// One wave per 16x16 output tile. Wave-uniform guard keeps EXEC all-1s.
__global__ void gemm_bias_wmma(const float* __restrict__ A, const float* __restrict__ W,
                               const float* __restrict__ bias, float* __restrict__ out,
                               int M, int K, int NC) {
  int wavesPerBlock = blockDim.x >> 5;
  int wave = blockIdx.x * wavesPerBlock + (threadIdx.x >> 5);
  int tilesN = NC >> 4;
  int tileM = wave / tilesN;
  int tileN = wave - tileM * tilesN;
  if ((tileM << 4) >= M) return;
  int lane = threadIdx.x & 31;
  int hh = lane >> 4;     // lane half selects K pairs (A) / rows (B,C)
  int l16 = lane & 15;
  int i0 = tileM << 4, j0 = tileN << 4;
  // A 16x4 layout: row = lane%16, K = vgpr + 2*(lane/16)
  const float* arow = A + (size_t)(i0 + l16) * K + 2 * hh;
  // B 4x16 layout: col = lane%16, K = vgpr + 2*(lane/16); B[k][n] = W[j0+n][k]
  const float* wrow = W + (size_t)(j0 + l16) * K + 2 * hh;
  float bv = bias[j0 + l16];
  v8f acc;
#pragma unroll
  for (int r = 0; r < 8; ++r) acc[r] = bv;  // bias broadcast along M
  for (int k0 = 0; k0 < K; k0 += 4) {
    v2f av, bvv;
    av.x = arow[k0];  av.y = arow[k0 + 1];
    bvv.x = wrow[k0]; bvv.y = wrow[k0 + 1];
    acc = __builtin_amdgcn_wmma_f32_16x16x4_f32(false, av, false, bvv,
                                                (short)0, acc, false, false);
  }
  // C/D 16x16 f32: row = r + 8*(lane/16), col = lane%16
#pragma unroll
  for (int r = 0; r < 8; ++r)
    out[(size_t)(i0 + r + 8 * hh) * NC + j0 + l16] = acc[r];
}

// PyTorch GRU gating: gates ordered (r, z, n)
__global__ void gru_gate_kernel(const float* __restrict__ gi, const float* __restrict__ gh,
                                float* __restrict__ h) {
  int idx = blockIdx.x * blockDim.x + threadIdx.x;
  if (idx >= BB * DD) return;
  int b = idx >> 7;      // / DD
  int d = idx & (DD - 1);
  const float* gib = gi + (size_t)b * 3 * DD;
  const float* ghb = gh + (size_t)b * 3 * DD;
  float r = 1.f / (1.f + expf(-(gib[d] + ghb[d])));
  float z = 1.f / (1.f + expf(-(gib[DD + d] + ghb[DD + d])));
  float nn = tanhf(gib[2 * DD + d] + r * ghb[2 * DD + d]);
  h[idx] = (1.f - z) * nn + z * h[idx];
}

// Fused attention: per-graph block. Phase 1: dot scores + online softmax stats
// (one HBM pass over this graph's x rows). Phase 2: weighted pooling, re-reading
// the same ~125 KB slice from L2. Writes q_star = [h | r_pool].
__global__ __launch_bounds__(128)
void attn_pool_kernel(const float* __restrict__ x, const int* __restrict__ seg,
                      const float* __restrict__ h, float* __restrict__ e,
                      float* __restrict__ dst /* [BB, 2*DD] */) {
  const int b = blockIdx.x;
  const int tid = threadIdx.x;
  const int n0 = seg[b], n1 = seg[b + 1];

  __shared__ __align__(16) float h_sh[DD];
  __shared__ float mg_sh[16], sg_sh[16];
  __shared__ float ms_sh[2];
  __shared__ float w_sh[128];

  h_sh[tid] = h[(size_t)b * DD + tid];
  __syncthreads();

  // ---- Phase 1: 16 groups of 8 lanes; each group handles one node at a time.
  const int g = tid >> 3, sub = tid & 7;
  float m_g = -3.402823466e38f, s_g = 0.f;
  const float4* hr = (const float4*)h_sh;
  for (int n = n0 + g; n < n1; n += 16) {
    const float4* xr = (const float4*)(x + (size_t)n * DD);
    float dot = 0.f;
#pragma unroll
    for (int i = 0; i < 4; ++i) {
      float4 xv = xr[sub + 8 * i];
      float4 hv = hr[sub + 8 * i];
      dot += xv.x * hv.x + xv.y * hv.y + xv.z * hv.z + xv.w * hv.w;
    }
    dot += __shfl_xor(dot, 1, 32);
    dot += __shfl_xor(dot, 2, 32);
    dot += __shfl_xor(dot, 4, 32);
    if (sub == 0) e[n] = dot;
    // online (max, sum-of-exp); state replicated across the 8 lanes
    if (dot > m_g) { s_g = s_g * expf(m_g - dot) + 1.f; m_g = dot; }
    else           { s_g += expf(dot - m_g); }
  }
  if (sub == 0) { mg_sh[g] = m_g; sg_sh[g] = s_g; }
  __syncthreads();
  if (tid == 0) {
    float m = -3.402823466e38f, s = 0.f;
    for (int i = 0; i < 16; ++i) {
      float mi = mg_sh[i], si = sg_sh[i];
      float nm = fmaxf(m, mi);
      s = s * expf(m - nm) + si * expf(mi - nm);
      m = nm;
    }
    ms_sh[0] = m; ms_sh[1] = s;
  }
  __syncthreads();
  const float m_tot = ms_sh[0];
  const float inv = 1.f / (ms_sh[1] + 1e-16f);   // EPS from reference

  // ---- Phase 2: chunks of 128 nodes; thread = feature dim.
  float acc = 0.f;
  for (int c = n0; c < n1; c += 128) {
    int cnt = min(128, n1 - c);
    __syncthreads();
    if (tid < cnt) w_sh[tid] = expf(e[c + tid] - m_tot);
    __syncthreads();
    for (int j = 0; j < cnt; ++j)
      acc += w_sh[j] * x[(size_t)(c + j) * DD + tid];
  }
  acc *= inv;
  dst[(size_t)b * 2 * DD + tid]      = h_sh[tid];  // q
  dst[(size_t)b * 2 * DD + DD + tid] = acc;        // r_pool
}

extern "C" void kernel_launch(void* const* d_in, const int* in_sizes, int n_in,
                              void* d_out, int out_size, void* d_ws, size_t ws_size,
                              hipStream_t stream) {
  (void)in_sizes; (void)n_in; (void)out_size; (void)ws_size;
  const float* x     = (const float*)d_in[0];
  const int*   batch = (const int*)d_in[1];
  const float* w_ih  = (const float*)d_in[2];
  const float* w_hh  = (const float*)d_in[3];
  const float* b_ih  = (const float*)d_in[4];
  const float* b_hh  = (const float*)d_in[5];
  float* out = (float*)d_out;

  // workspace layout (~23 MB)
  float* h     = (float*)d_ws;          // BB*DD
  float* qstar = h + BB * DD;           // BB*2*DD
  float* gi    = qstar + BB * 2 * DD;   // BB*3*DD
  float* gh    = gi + BB * 3 * DD;      // BB*3*DD
  float* e     = gh + BB * 3 * DD;      // NB
  int*   seg   = (int*)(e + NB);        // BB+1

  // zero h and q_star (contiguous)
  int zc = BB * DD + BB * 2 * DD;
  zero_kernel<<<(zc + 255) / 256, 256, 0, stream>>>(h, zc);
  seg_bounds_kernel<<<(NB + 1 + 255) / 256, 256, 0, stream>>>(batch, seg);

  for (int t = 0; t < TT; ++t) {
    // gi = q_star @ w_ih^T + b_ih   (4096x256 * 256x384)
    gemm_bias_wmma<<<768, 256, 0, stream>>>(qstar, w_ih, b_ih, gi, BB, 2 * DD, 3 * DD);
    // gh = h @ w_hh^T + b_hh        (4096x128 * 128x384)
    gemm_bias_wmma<<<768, 256, 0, stream>>>(h, w_hh, b_hh, gh, BB, DD, 3 * DD);
    gru_gate_kernel<<<(BB * DD + 255) / 256, 256, 0, stream>>>(gi, gh, h);
    attn_pool_kernel<<<BB, 128, 0, stream>>>(x, seg, h, e,
                                             (t == TT - 1) ? out : qstar);
  }
}